// Attention_65343632441735
// MI455X (gfx1250) — compile-verified
//
#include <hip/hip_runtime.h>
#include <math.h>
#include <stdint.h>

typedef float v2f __attribute__((ext_vector_type(2)));
typedef float v8f __attribute__((ext_vector_type(8)));
typedef unsigned int u32x4 __attribute__((ext_vector_type(4)));
typedef int i32x4 __attribute__((ext_vector_type(4)));
typedef int i32x8 __attribute__((ext_vector_type(8)));

#define HW     16384
#define KDIM   192
#define CHUNKS 32
#define CHUNK_PIX (HW / CHUNKS)   // 512 pixels -> 128 k-steps per wave
#define LROW   132                // 96 dwords + 3*12 pad dwords (TDM pad_enable)

// ---------------------------------------------------------------------------
// TDM: DMA one A-panel (96 rows x 96 K, fp32, global row stride 192) into LDS
// with 12-dword padding every 32 dwords -> LDS row stride 132 dwords
// (132 mod 64 == 4  =>  conflict-free 16-row x 4-col wave reads).
// D# packed per CDNA5 ISA.
// ---------------------------------------------------------------------------
__device__ inline void tdm_load_panel(const float* gsrc, unsigned lds_byte_off)
{
    unsigned long long ga = (unsigned long long)(uintptr_t)gsrc;
    u32x4 g0;
    g0.x = 1u;                                            // count=1 (valid, user)
    g0.y = lds_byte_off;                                  // lds_addr (bytes)
    g0.z = (unsigned)(ga & 0xFFFFFFFFull);                // global_addr[31:0]
    g0.w = (unsigned)((ga >> 32) & 0x1FFFFFFull)          // global_addr[56:32]
         | (2u << 30);                                    // type=2 ("image")

    i32x8 g1;
    g1[0] = (int)0x17120000u;  // data_size=4B | pad_en | pad_interval=32dw | pad=12dw
    g1[1] = (int)(192u << 16); // tensor_dim0 = 192 (row length for OOB)
    g1[2] = (int)(96u << 16);  // tensor_dim1 = 96
    g1[3] = (int)(96u << 16);  // tile_dim0   = 96 dwords per row
    g1[4] = 96;                // tile_dim1   = 96 rows, tile_dim2 = 0
    g1[5] = 192;               // tensor_dim0_stride = 192 elements
    g1[6] = 0;
    g1[7] = 0;

    i32x4 z4 = {0, 0, 0, 0};
#if __clang_major__ >= 23
    i32x8 z8 = {0, 0, 0, 0, 0, 0, 0, 0};
    __builtin_amdgcn_tensor_load_to_lds(g0, g1, z4, z4, z8, 0);
#else
    __builtin_amdgcn_tensor_load_to_lds(g0, g1, z4, z4, 0);
#endif
}

// ---------------------------------------------------------------------------
// GEMM: C[M x HW] = A[M x 192] * B[192 x HW], optionally B = B1 + B2 (x+f fuse)
// fp32 WMMA 16x16x4. Wave: 6 M-tiles x 1 N-tile; 8 waves/block share the
// TDM-staged A panel in LDS. K processed in two 96-wide stages.
// ---------------------------------------------------------------------------
template <bool HAS_B2>
__global__ __launch_bounds__(256)
void gemm192_wmma(const float* __restrict__ A, long strideA,
                  const float* __restrict__ B1, const float* __restrict__ B2,
                  long strideB,
                  float* __restrict__ C, long strideC)
{
    __shared__ float ldsA[96 * LROW];   // 50688 B

    const int lane   = threadIdx.x & 31;
    const int wave   = threadIdx.x >> 5;
    const int batch  = blockIdx.z;
    const int ntile  = blockIdx.x * 8 + wave;
    const int mtile0 = blockIdx.y * 6;

    const float* Ab  = A + (long)batch * strideA + (long)mtile0 * 16 * KDIM;
    const float* B1b = B1 + (long)batch * strideB;
    const float* B2b = HAS_B2 ? (B2 + (long)batch * strideB) : nullptr;
    float*       Cb  = C + (long)batch * strideC;

    const int m  = lane & 15;           // A row within tile / B,C column
    const int kb = (lane >> 4) * 2;     // K sub-offset for this half-wave
    const int p  = ntile * 16 + m;      // global pixel (N) index

    v8f acc[6] = {};

    for (int ks = 0; ks < KDIM; ks += 96) {
        if (threadIdx.x == 0)
            tdm_load_panel(Ab + ks, (unsigned)(uintptr_t)(void*)ldsA);
        if (threadIdx.x < 32)
            __builtin_amdgcn_s_wait_tensorcnt(0);
        __syncthreads();                 // panel visible to all 8 waves

        for (int kk = 0; kk < 96; kk += 4) {
            const int kg = ks + kk + kb;
            float b0 = B1b[(long)kg * HW + p];
            float b1 = B1b[(long)(kg + 1) * HW + p];
            if (HAS_B2) {
                b0 += B2b[(long)kg * HW + p];
                b1 += B2b[(long)(kg + 1) * HW + p];
            }
            v2f bm; bm.x = b0; bm.y = b1;
            const int cidx = kk + kb + 12 * (kk >> 5);   // padded column index
#pragma unroll
            for (int t = 0; t < 6; ++t) {
                const v2f am = *(const v2f*)(ldsA + (t * 16 + m) * LROW + cidx);
                acc[t] = __builtin_amdgcn_wmma_f32_16x16x4_f32(
                    false, am, false, bm, (short)0, acc[t], false, false);
            }
        }
        __syncthreads();                 // all reads done before next TDM stage
    }

    const int rb = (lane >> 4) * 8;
#pragma unroll
    for (int t = 0; t < 6; ++t) {
#pragma unroll
        for (int i = 0; i < 8; ++i) {
            const int row = (mtile0 + t) * 16 + rb + i;
            Cb[(long)row * HW + p] = acc[t][i];
        }
    }
}

// ---------------------------------------------------------------------------
// Depthwise 3x3 (SAME) fused with q/k sum-of-squares partial reduction.
// Block = 256 contiguous pixels of one (b, ch) row; one atomicAdd per block.
// ---------------------------------------------------------------------------
__global__ __launch_bounds__(256)
void dwconv3_sumsq(const float* __restrict__ in, const float* __restrict__ wdw,
                   float* __restrict__ out, float* __restrict__ norms)
{
    __shared__ float red[256];
    const int blk   = blockIdx.x;          // (b*576 + ch)*64 + chunk
    const int chunk = blk & 63;
    const int bc    = blk >> 6;            // b*576 + ch
    const int ch    = bc % 576;
    const int p     = chunk * 256 + threadIdx.x;
    const int x = p & 127, y = p >> 7;

    const float* ib = in + (long)bc * HW;
    const float* w  = wdw + ch * 9;
    float s = 0.f;
#pragma unroll
    for (int dy = -1; dy <= 1; ++dy) {
        int yy = y + dy;
        if (yy < 0 || yy > 127) continue;
#pragma unroll
        for (int dx = -1; dx <= 1; ++dx) {
            int xx = x + dx;
            if (xx < 0 || xx > 127) continue;
            s += w[(dy + 1) * 3 + (dx + 1)] * ib[yy * 128 + xx];
        }
    }
    out[(long)bc * HW + p] = s;

    if (ch < 384) {                        // q or k channel: accumulate ||.||^2
        red[threadIdx.x] = s * s;
        __syncthreads();
        for (int st = 128; st > 0; st >>= 1) {
            if (threadIdx.x < st) red[threadIdx.x] += red[threadIdx.x + st];
            __syncthreads();
        }
        if (threadIdx.x == 0)
            atomicAdd(&norms[(bc / 576) * 384 + ch], red[0]);
    }
}

// ---------------------------------------------------------------------------
// Raw Gram matrix: attn[b,head] += Q_chunk * K_chunk^T  (48x48, WMMA f32)
// One wave per (b*head, hw-chunk); 3x3 tile block; atomic accumulate.
// ---------------------------------------------------------------------------
__global__ __launch_bounds__(32)
void attn_qk_wmma(const float* __restrict__ qkv, float* __restrict__ attn)
{
    const int lane  = threadIdx.x & 31;
    const int chunk = blockIdx.x;
    const int bh    = blockIdx.y;          // b*4 + head
    const int b = bh >> 2, head = bh & 3;
    const float* Q  = qkv + ((long)b * 576 + head * 48) * HW;
    const float* Kp = qkv + ((long)b * 576 + 192 + head * 48) * HW;

    const int m  = lane & 15;
    const int kb = (lane >> 4) * 2;

    v8f acc[3][3] = {};

    for (int s = 0; s < CHUNK_PIX; s += 4) {
        const int k = chunk * CHUNK_PIX + s + kb;     // even -> 8B aligned
        v2f a[3], bm[3];
#pragma unroll
        for (int mt = 0; mt < 3; ++mt)
            a[mt] = *(const v2f*)(Q + (long)(mt * 16 + m) * HW + k);
#pragma unroll
        for (int nt = 0; nt < 3; ++nt)
            bm[nt] = *(const v2f*)(Kp + (long)(nt * 16 + m) * HW + k);
#pragma unroll
        for (int mt = 0; mt < 3; ++mt)
#pragma unroll
            for (int nt = 0; nt < 3; ++nt)
                acc[mt][nt] = __builtin_amdgcn_wmma_f32_16x16x4_f32(
                    false, a[mt], false, bm[nt], (short)0, acc[mt][nt], false, false);
    }

    float* base = attn + (long)bh * 48 * 48;
    const int col = lane & 15;
    const int rb  = (lane >> 4) * 8;
#pragma unroll
    for (int mt = 0; mt < 3; ++mt)
#pragma unroll
        for (int nt = 0; nt < 3; ++nt)
#pragma unroll
            for (int i = 0; i < 8; ++i)
                atomicAdd(&base[(mt * 16 + rb + i) * 48 + nt * 16 + col],
                          acc[mt][nt][i]);
}

// ---------------------------------------------------------------------------
// Fold temperature and q/k L2 norms into attn, softmax over last dim (48).
// ---------------------------------------------------------------------------
__global__ void softmax_rows(float* __restrict__ attn,
                             const float* __restrict__ norms,
                             const float* __restrict__ temp)
{
    int r = blockIdx.x * blockDim.x + threadIdx.x;
    if (r >= 8 * 4 * 48) return;
    int c = r % 48; int bh = r / 48; int head = bh & 3; int b = bh >> 2;
    const float eps = 1e-12f;
    float nq = fmaxf(sqrtf(norms[b * 384 + head * 48 + c]), eps);
    float t = temp[head];
    float* row = attn + (long)bh * 48 * 48 + c * 48;
    float vals[48];
    float mx = -3.0e38f;
    for (int d = 0; d < 48; ++d) {
        float nk = fmaxf(sqrtf(norms[b * 384 + 192 + head * 48 + d]), eps);
        float v = row[d] * t / (nq * nk);
        vals[d] = v; mx = fmaxf(mx, v);
    }
    float sum = 0.f;
    for (int d = 0; d < 48; ++d) { vals[d] = __expf(vals[d] - mx); sum += vals[d]; }
    float inv = 1.f / sum;
    for (int d = 0; d < 48; ++d) row[d] = vals[d] * inv;
}

// ---------------------------------------------------------------------------
// M_b[co, dg] = sum_cl Wproj[co, head(dg)*48+cl] * attn[b, head, cl, dl]
// (Wproj @ BlockDiag(attn_b)) -- fuses attn@v with the output projection.
// ---------------------------------------------------------------------------
__global__ void build_m(const float* __restrict__ wproj,
                        const float* __restrict__ attn,
                        float* __restrict__ M)
{
    int idx = blockIdx.x * blockDim.x + threadIdx.x;
    if (idx >= 8 * 192 * 192) return;
    int dg = idx % 192; int t = idx / 192; int co = t % 192; int b = t / 192;
    int head = dg / 48, dl = dg % 48;
    const float* arow = attn + (long)(b * 4 + head) * 48 * 48 + dl;
    const float* wrow = wproj + co * 192 + head * 48;
    float s = 0.f;
    for (int cl = 0; cl < 48; ++cl) s += wrow[cl] * arow[cl * 48];
    M[idx] = s;
}

// ---------------------------------------------------------------------------
extern "C" void kernel_launch(void* const* d_in, const int* in_sizes, int n_in,
                              void* d_out, int out_size, void* d_ws, size_t ws_size,
                              hipStream_t stream)
{
    (void)in_sizes; (void)n_in; (void)out_size; (void)ws_size;
    const float* x     = (const float*)d_in[0];
    const float* f     = (const float*)d_in[1];
    const float* wqkv  = (const float*)d_in[2];
    const float* wdw   = (const float*)d_in[3];
    const float* wproj = (const float*)d_in[4];
    const float* temp  = (const float*)d_in[5];
    float* out = (float*)d_out;

    float* qkv_pre = (float*)d_ws;                         // 8*576*HW
    float* qkv_dw  = qkv_pre + (size_t)8 * 576 * HW;       // 8*576*HW
    float* norms   = qkv_dw  + (size_t)8 * 576 * HW;       // 8*384
    float* attn    = norms   + 8 * 384;                    // 8*4*48*48
    float* Mmat    = attn    + 8 * 4 * 48 * 48;            // 8*192*192

    // zero norm + attn accumulators (contiguous)
    hipMemsetAsync(norms, 0,
                   (size_t)(8 * 384 + 8 * 4 * 48 * 48) * sizeof(float), stream);

    // 1) qkv = Wqkv @ (x + f)   [per batch: 576x192 @ 192x16384]
    dim3 g1(HW / 16 / 8, 36 / 6, 8);
    gemm192_wmma<true><<<g1, 256, 0, stream>>>(wqkv, 0L, x, f, (long)192 * HW,
                                               qkv_pre, (long)576 * HW);

    // 2) depthwise 3x3 fused with q/k sum-of-squares
    dwconv3_sumsq<<<8 * 576 * 64, 256, 0, stream>>>(qkv_pre, wdw, qkv_dw, norms);

    // 3) raw Gram matrices (split-K atomic accumulate)
    dim3 g5(CHUNKS, 32);
    attn_qk_wmma<<<g5, 32, 0, stream>>>(qkv_dw, attn);

    // 4) scale by temperature / (||q|| ||k||), softmax over 48
    softmax_rows<<<(1536 + 63) / 64, 64, 0, stream>>>(attn, norms, temp);

    // 5) M_b = Wproj @ BlockDiag(attn_b)
    int tm = 8 * 192 * 192;
    build_m<<<(tm + 255) / 256, 256, 0, stream>>>(wproj, attn, Mmat);

    // 6) out = M_b @ V   [per batch: 192x192 @ 192x16384]
    dim3 g8(HW / 16 / 8, 12 / 6, 8);
    gemm192_wmma<false><<<g8, 256, 0, stream>>>(Mmat, (long)192 * 192,
                                                qkv_dw + (size_t)384 * HW, nullptr,
                                                (long)576 * HW, out, (long)192 * HW);
}